// SE2NonLinearity_42125039239524
// MI455X (gfx1250) — compile-verified
//
#include <hip/hip_runtime.h>
#include <math.h>

// SE2 nonlinearity: per (batch, column) 16-point unitary IDFT -> ReLU(re/im) -> DFT.
// Memory bound (~537 MB @ 23.3 TB/s). FP32 WMMA (V_WMMA_F32_16X16X4_F32) keeps
// full precision; complex matmul decomposed into real 16x16x16 WMMA chains.
// v3: zero-shuffle chaining — stage-2 consumes stage-1's D-layout registers
//     directly as B operands; the row permutation is folded into the stage-2
//     twiddle (A) tables, eliminating all 32 ds_bpermute per tile.

typedef float v2f __attribute__((ext_vector_type(2)));
typedef float v8f __attribute__((ext_vector_type(8)));

#define NCOLS 131072L          // 32*64*64 columns per (batch, m) row
#define TILES_PER_BATCH 8192L  // NCOLS / 16
#define TILES_PER_WAVE 8

static __device__ inline v8f wmma4(v2f a, v2f b, v8f c) {
  // D(16x16,f32) = A(16x4,f32) * B(4x16,f32) + C
  return __builtin_amdgcn_wmma_f32_16x16x4_f32(
      /*neg_a=*/false, a, /*neg_b=*/false, b,
      /*c_mod=*/(short)0, c, /*reuse_a=*/false, /*reuse_b=*/false);
}

__global__ __launch_bounds__(256, 1) void se2_dft_relu_dft(
    const float* __restrict__ xr, const float* __restrict__ xi,
    float* __restrict__ out) {
  const int lane = threadIdx.x & 31;
  const int arow = lane & 15;   // A row / B,C,D column index
  const int hi   = lane >> 4;   // 0: lanes 0-15, 1: lanes 16-31
  const long wave = (long)blockIdx.x * (blockDim.x >> 5) + (threadIdx.x >> 5);

  // ---- Twiddle tables (one-time, native v_cos/v_sin).
  // A(16x4,f32) layout: lane (arow,hi) element e in {x,y} holds K = 4q + 2*hi + e.
  //
  // Stage 1 (B loaded from memory in canonical B layout): column index k1 = K.
  // Stage 2 (B = stage-1 D registers passed through unpermuted): D slot
  //   (v=2q+p, half h) holds row 2q+p+8h but is read as K=4q+p+2h, so the A
  //   table must hold W[arow][pi(K)] with pi(4q+p+2h) = 2q+p+8h, i.e. element
  //   e of chunk q uses column k2 = 2q + 8*hi + e.
  v2f Ac1[4], As1[4], Ans1[4];   // stage-1: cos, sin, -sin
  v2f Ac2[4], As2[4], Ans2[4];   // stage-2: permuted cos, sin, -sin
#pragma unroll
  for (int q = 0; q < 4; ++q) {
    const int k1 = 4 * q + 2 * hi;
    const int k2 = 2 * q + 8 * hi;
    const float c = 3.14159265358979323846f / 8.0f;
    const float a10 = (float)((arow * k1) & 15) * c;
    const float a11 = (float)((arow * (k1 + 1)) & 15) * c;
    const float a20 = (float)((arow * k2) & 15) * c;
    const float a21 = (float)((arow * (k2 + 1)) & 15) * c;
    Ac1[q].x = __cosf(a10);  Ac1[q].y = __cosf(a11);
    As1[q].x = __sinf(a10);  As1[q].y = __sinf(a11);
    Ans1[q].x = -As1[q].x;   Ans1[q].y = -As1[q].y;
    Ac2[q].x = __cosf(a20);  Ac2[q].y = __cosf(a21);
    As2[q].x = __sinf(a20);  As2[q].y = __sinf(a21);
    Ans2[q].x = -As2[q].x;   Ans2[q].y = -As2[q].y;
  }

  v2f* outv = reinterpret_cast<v2f*>(out);

  for (int it = 0; it < TILES_PER_WAVE; ++it) {
    const long tile = wave * TILES_PER_WAVE + it;   // 131072 tiles total
    const long b = tile >> 13;                      // tile / TILES_PER_BATCH
    const long n = (tile & (TILES_PER_BATCH - 1)) * 16 + arow;

    // ---- Load X tile as B operands (K-major chunks of 4 rows), non-temporal.
    // B(4x16) layout: VGPR p, lanes0-15 -> K=4q+p, lanes16-31 -> K=4q+p+2.
    v2f Bxr[4], Bxi[4];
#pragma unroll
    for (int q = 0; q < 4; ++q) {
      const long r0 = (long)(q * 4 + hi * 2);
      const long i0 = (b * 16 + r0) * NCOLS + n;
      Bxr[q].x = __builtin_nontemporal_load(xr + i0);
      Bxr[q].y = __builtin_nontemporal_load(xr + i0 + NCOLS);
      Bxi[q].x = __builtin_nontemporal_load(xi + i0);
      Bxi[q].y = __builtin_nontemporal_load(xi + i0 + NCOLS);
    }

    // ---- Stage 1: inverse DFT (unscaled; unitary scales commute with ReLU).
    // Yr = c*Xr - s*Xi ; Yi = c*Xi + s*Xr
    v8f Yr = {}, Yi = {};
#pragma unroll
    for (int q = 0; q < 4; ++q) Yr = wmma4(Ac1[q], Bxr[q], Yr);
#pragma unroll
    for (int q = 0; q < 4; ++q) Yr = wmma4(Ans1[q], Bxi[q], Yr);
#pragma unroll
    for (int q = 0; q < 4; ++q) Yi = wmma4(Ac1[q], Bxi[q], Yi);
#pragma unroll
    for (int q = 0; q < 4; ++q) Yi = wmma4(As1[q], Bxr[q], Yi);

    // ---- ReLU, then repack D registers as stage-2 B chunks (pure reg copies;
    //      the row permutation is compensated in Ac2/As2/Ans2).
    v2f Byr[4], Byi[4];
#pragma unroll
    for (int q = 0; q < 4; ++q) {
      Byr[q].x = fmaxf(Yr[2 * q], 0.0f);
      Byr[q].y = fmaxf(Yr[2 * q + 1], 0.0f);
      Byi[q].x = fmaxf(Yi[2 * q], 0.0f);
      Byi[q].y = fmaxf(Yi[2 * q + 1], 0.0f);
    }

    // ---- Stage 2: forward DFT. Zr = c*Yr + s*Yi ; Zi = c*Yi - s*Yr
    v8f Zr = {}, Zi = {};
#pragma unroll
    for (int q = 0; q < 4; ++q) Zr = wmma4(Ac2[q], Byr[q], Zr);
#pragma unroll
    for (int q = 0; q < 4; ++q) Zr = wmma4(As2[q], Byi[q], Zr);
#pragma unroll
    for (int q = 0; q < 4; ++q) Zi = wmma4(Ac2[q], Byi[q], Zi);
#pragma unroll
    for (int q = 0; q < 4; ++q) Zi = wmma4(Ans2[q], Byr[q], Zi);

    // ---- Apply combined unitary scale (1/64 * 1/4), store interleaved re/im
    //      as packed b64 non-temporal stores.
    const float sc = 1.0f / 256.0f;
    const long obase = (b * 16 + (long)hi * 8) * NCOLS + n;  // v2f units
#pragma unroll
    for (int v = 0; v < 8; ++v) {
      v2f z;
      z.x = Zr[v] * sc;
      z.y = Zi[v] * sc;
      __builtin_nontemporal_store(z, outv + obase + (long)v * NCOLS);
    }
  }
}

extern "C" void kernel_launch(void* const* d_in, const int* in_sizes, int n_in,
                              void* d_out, int out_size, void* d_ws, size_t ws_size,
                              hipStream_t stream) {
  (void)in_sizes; (void)n_in; (void)out_size; (void)d_ws; (void)ws_size;
  const float* xr = (const float*)d_in[0];
  const float* xi = (const float*)d_in[1];
  float* out = (float*)d_out;
  // 131072 tiles / 8 tiles-per-wave = 16384 waves; 8 waves (256 thr) per block.
  dim3 grid(2048), block(256);
  hipLaunchKernelGGL(se2_dft_relu_dft, grid, block, 0, stream, xr, xi, out);
}